// snn_layer_conv_16947940950664
// MI455X (gfx1250) — compile-verified
//
#include <hip/hip_runtime.h>

typedef float v2f  __attribute__((ext_vector_type(2)));
typedef float v8f  __attribute__((ext_vector_type(8)));
typedef unsigned int u32x4 __attribute__((ext_vector_type(4)));
typedef int   i32x4 __attribute__((ext_vector_type(4)));
typedef int   i32x8 __attribute__((ext_vector_type(8)));

#define HH   512
#define WW   512
#define NB   64
#define OH   510
#define OW   510
#define NSTEPS 1000
#define BX   128            // output cols per block (64 threads x 2)
#define BY   4              // output rows per block
#define TW   (BX + 2)       // 130 tile width (floats)
#define THR  (BY + 2)       // 6 tile rows

#if defined(__gfx1250__)
#if __has_builtin(__builtin_amdgcn_tensor_load_to_lds) && __has_builtin(__builtin_amdgcn_s_wait_tensorcnt)
#define USE_TDM 1
#endif
#if __has_builtin(__builtin_amdgcn_wmma_f32_16x16x4_f32)
#define USE_WMMA 1
#endif
#endif

__global__ __launch_bounds__(256) void snn_layer_conv_kernel(const float* __restrict__ in,
                                                             float* __restrict__ out)
{
    __shared__ float tile[THR * TW];   // 6 x 130 input tile (row stride 130, TDM-packed)
    __shared__ float ysum[BY * BX];    // conv results, element id = ty*128 + col_local

    const int tx  = threadIdx.x;       // 0..63
    const int ty  = threadIdx.y;       // 0..3
    const int tid = ty * 64 + tx;
    const int x0  = blockIdx.x * BX;
    const int r0  = blockIdx.y * BY;
    const int b   = blockIdx.z;

    const long long row0g = (long long)b * HH + r0;          // global row of tile origin
    const float*    gbase = in + ((size_t)row0g * WW + x0);  // tile origin in memory
    const int       tdx   = WW - x0;                         // valid cols from tile origin
    const long long tdy   = (long long)NB * HH - row0g;      // valid rows from tile origin

    // ---------------- Stage input tile into LDS ----------------
#ifdef USE_TDM
    if (ty == 0 && tx < 32) {   // wave 0 issues the TDM descriptor
        unsigned long long ga = (unsigned long long)(size_t)gbase;
        u32x4 g0;
        g0.x = 1u;                                          // count=1, user desc
        g0.y = (unsigned)(size_t)(&tile[0]);                // lds_addr (bytes)
        g0.z = (unsigned)(ga & 0xFFFFFFFFull);              // global_addr[31:0]
        g0.w = (unsigned)((ga >> 32) & 0x1FFFFFFull) | (2u << 30); // addr[56:32] | type=2
        i32x8 g1;
        g1[0] = 0x00020000;                                 // data_size = 4 bytes
        g1[1] = (int)((tdx & 0xFFFF) << 16);                // tensor_dim0 lo16
        g1[2] = (int)(((tdx >> 16) & 0xFFFF) | (int)((tdy & 0xFFFFll) << 16)); // dim0 hi | dim1 lo
        g1[3] = (int)(((int)(tdy >> 16) & 0xFFFF) | (TW << 16)); // dim1 hi | tile_dim0 = 130
        g1[4] = THR;                                        // tile_dim1 = 6
        g1[5] = WW;                                         // tensor_dim0_stride lo32 = 512
        g1[6] = 0;
        g1[7] = 0;
        i32x4 gz = {0, 0, 0, 0};
#if __clang_major__ >= 23
        i32x8 gz8 = {0, 0, 0, 0, 0, 0, 0, 0};
        __builtin_amdgcn_tensor_load_to_lds(g0, g1, gz, gz, gz8, 0);
#else
        __builtin_amdgcn_tensor_load_to_lds(g0, g1, gz, gz, 0);
#endif
        __builtin_amdgcn_s_wait_tensorcnt(0);
    }
#else
    for (int i = tid; i < THR * TW; i += 256) {
        int rr = i / TW;
        int cc = i - rr * TW;
        float v = 0.0f;                                     // mirror TDM OOB zero-fill
        if (cc < tdx && rr < tdy) v = gbase[(size_t)rr * WW + cc];
        tile[i] = v;
    }
#endif
    __syncthreads();

    // ---------------- 3x3 box-filter (conv with ones kernel) ----------------
    float y0, y1;
#ifdef USE_WMMA
    {
        // Exact f32 WMMA: D = A x B + C with B == all-ones (products exact,
        // immune to B-layout permutation). A is 16x4 f32 per documented layout:
        // lanes 0-15 supply K=0,1 for row M=lane; lanes 16-31 supply K=2,3.
        // 3 chained WMMAs give K=0..11; slots 9..11 are zero (9-point window).
        const int lane = tx & 31;
        const int half = lane >> 4;
        const int m    = lane & 15;
        const int tx0  = tx & 32;                 // wave-uniform column base
        v2f bones; bones[0] = 1.0f; bones[1] = 1.0f;
        #pragma unroll
        for (int q = 0; q < 4; ++q) {
            const int clbase = tx0 + (q & 1) * 16 + (q >> 1) * 64;
            v8f acc = {0.0f, 0.0f, 0.0f, 0.0f, 0.0f, 0.0f, 0.0f, 0.0f};
            #pragma unroll
            for (int c = 0; c < 3; ++c) {
                int k0 = 4 * c + 2 * half;
                int k1 = k0 + 1;
                float a0e = (k0 <= 8) ? tile[(ty + k0 / 3) * TW + clbase + m + (k0 % 3)] : 0.0f;
                float a1e = (k1 <= 8) ? tile[(ty + k1 / 3) * TW + clbase + m + (k1 % 3)] : 0.0f;
                v2f av; av[0] = a0e; av[1] = a1e;
                acc = __builtin_amdgcn_wmma_f32_16x16x4_f32(
                          false, av, false, bones, (short)0, acc, false, false);
            }
            // D layout: VGPR v -> M=v (lanes 0-15) / M=v+8 (lanes 16-31); N=lane (all equal).
            if (m == 0) {
                int ebase = ty * BX + clbase + half * 8;
                #pragma unroll
                for (int v = 0; v < 8; ++v) ysum[ebase + v] = acc[v];
            }
        }
        __syncthreads();
        y0 = ysum[ty * BX + tx];
        y1 = ysum[ty * BX + tx + 64];
    }
#else
    {
        y0 = 0.0f; y1 = 0.0f;
        #pragma unroll
        for (int dr = 0; dr < 3; ++dr)
            #pragma unroll
            for (int dc = 0; dc < 3; ++dc) {
                y0 += tile[(ty + dr) * TW + tx + dc];
                y1 += tile[(ty + dr) * TW + tx + 64 + dc];
            }
    }
#endif

    // ---------------- 1000-step LIF recurrence (VALU-bound hot loop) ----------------
    // step3(n) o step1(n+1) fuses exactly to: V<14 -> V ; V==14 -> 14 ; V>14 -> 0.
    const float cstep = (float)(0.01 / 30000.0);     // DT / (R*C), rounded once like JAX
    const float Ry0 = __fmul_rn(3000.0f, y0);
    const float Ry1 = __fmul_rn(3000.0f, y1);
    float V0 = 0.0f, V1 = 0.0f;
    #pragma unroll 4
    for (int n = 0; n < NSTEPS; ++n) {
        float t0 = (V0 > 14.0f) ? 0.0f : 14.0f;
        float t1 = (V1 > 14.0f) ? 0.0f : 14.0f;
        V0 = (V0 < 14.0f) ? V0 : t0;
        V1 = (V1 < 14.0f) ? V1 : t1;
        // V + (-V + R*y) * (DT/TM), non-contracted to match the literal formula
        V0 = __fadd_rn(V0, __fmul_rn(__fadd_rn(-V0, Ry0), cstep));
        V1 = __fadd_rn(V1, __fmul_rn(__fadd_rn(-V1, Ry1), cstep));
    }
    // final spike shaping: V<14 -> V ; V==14 -> 14 ; V>14 -> 28
    float s0 = (V0 > 14.0f) ? 28.0f : 14.0f;
    float s1 = (V1 > 14.0f) ? 28.0f : 14.0f;
    V0 = (V0 < 14.0f) ? V0 : s0;
    V1 = (V1 < 14.0f) ? V1 : s1;

    // ---------------- store ----------------
    const int oy = r0 + ty;
    const int c0 = x0 + tx;
    const int c1 = c0 + 64;
    if (oy < OH) {
        size_t rowo = ((size_t)b * OH + oy) * OW;
        if (c0 < OW) out[rowo + c0] = V0;
        if (c1 < OW) out[rowo + c1] = V1;
    }
}

extern "C" void kernel_launch(void* const* d_in, const int* in_sizes, int n_in,
                              void* d_out, int out_size, void* d_ws, size_t ws_size,
                              hipStream_t stream) {
    (void)in_sizes; (void)n_in; (void)out_size; (void)d_ws; (void)ws_size;
    const float* in  = (const float*)d_in[0];
    float*       out = (float*)d_out;
    dim3 block(64, 4, 1);                                   // 256 threads = 8 waves
    dim3 grid((OW + BX - 1) / BX, (OH + BY - 1) / BY, NB);  // 4 x 128 x 64 blocks
    hipLaunchKernelGGL(snn_layer_conv_kernel, grid, block, 0, stream, in, out);
}